// NodeBlock_89163521065282
// MI455X (gfx1250) — compile-verified
//
#include <hip/hip_runtime.h>

typedef __attribute__((ext_vector_type(2))) float v2f;
typedef __attribute__((ext_vector_type(8))) float v8f;

#define N_NODES 50000
#define N_EDGES 800000
#define D 128
#define HID 128
#define TILE 16

// LDS strides in floats, padded so WMMA A-operand reads are bank-conflict-free
#define CAT_STRIDE 260   // 256 + 4  -> bank = (4*row + k) % 64, rows distinct
#define H_STRIDE   132   // 128 + 4

__global__ void zero_kernel(float4* __restrict__ p, int n4) {
  int i = blockIdx.x * blockDim.x + threadIdx.x;
  if (i < n4) p[i] = make_float4(0.f, 0.f, 0.f, 0.f);
}

// one wave32 per edge: each lane loads float4 (512B coalesced / wave) and
// scatter-adds into the L2-resident agg accumulator
__global__ void scatter_kernel(const float* __restrict__ efeat,
                               const int* __restrict__ dst_idx,
                               float* __restrict__ agg) {
  int gid  = blockIdx.x * blockDim.x + threadIdx.x;
  int edge = gid >> 5;
  int lane = gid & 31;
  if (edge >= N_EDGES) return;
  // edge is uniform across the wave: force a scalar load + broadcast
  int d = __builtin_amdgcn_readfirstlane(dst_idx[edge]);
  float4 v = ((const float4*)(efeat + (size_t)edge * D))[lane];
  float* out = agg + (size_t)d * D + (size_t)lane * 4;
  atomicAdd(out + 0, v.x);
  atomicAdd(out + 1, v.y);
  atomicAdd(out + 2, v.z);
  atomicAdd(out + 3, v.w);
}

__global__ __launch_bounds__(256, 2)
void mlp_kernel(const float* __restrict__ agg,
                const float* __restrict__ nfeat,
                const float* __restrict__ w1, const float* __restrict__ b1,
                const float* __restrict__ w2, const float* __restrict__ b2,
                const float* __restrict__ ln_g, const float* __restrict__ ln_b,
                float* __restrict__ out) {
  __shared__ float cat_lds[TILE * CAT_STRIDE];  // reused as out_lds after GEMM1
  __shared__ float h_lds[TILE * H_STRIDE];

  const int tid   = threadIdx.x;
  const int wave  = tid >> 5;       // 0..7
  const int lane  = tid & 31;
  const int hi    = lane >> 4;      // half-wave select (K split for f32 WMMA)
  const int l16   = lane & 15;
  const int tile0 = blockIdx.x * TILE;

  // ---- stage cat = [agg | nfeat] (16 x 256) into LDS ----
  {
    int row  = tid >> 4;            // 0..15
    int cb   = (tid & 15) * 16;     // column base (multiple of 16)
    int node = tile0 + row;
    const float* src = (cb < D) ? (agg   + (size_t)node * D + cb)
                                : (nfeat + (size_t)node * D + (cb - D));
    float* dstp = cat_lds + row * CAT_STRIDE + cb;
#pragma unroll
    for (int i = 0; i < 4; ++i)
      *(float4*)(dstp + 4 * i) = *(const float4*)(src + 4 * i);
  }
  __syncthreads();

  const int n0   = wave * 16;       // this wave's output-column tile
  const int ncol = n0 + l16;        // C/D layout: col = lane&15

  // ---- GEMM1: h = silu(cat @ w1 + b1), K = 256, fp32 WMMA 16x16x4 ----
  v8f acc;
  {
    float bias = b1[ncol];
#pragma unroll
    for (int j = 0; j < 8; ++j) acc[j] = bias;
  }
  const float* arow = cat_lds + l16 * CAT_STRIDE;   // A: M = lane&15
#pragma unroll 4
  for (int kt = 0; kt < (2 * D) / 4; ++kt) {
    int k = kt * 4 + hi * 2;        // lanes 0-15: K=k0,k0+1; 16-31: K=k0+2,k0+3
    v2f a, b;
    a.x = arow[k];
    a.y = arow[k + 1];
    b.x = w1[(size_t)k * HID + ncol];
    b.y = w1[(size_t)(k + 1) * HID + ncol];
    acc = __builtin_amdgcn_wmma_f32_16x16x4_f32(false, a, false, b,
                                                (short)0, acc, false, false);
  }
  // SiLU via v_exp_f32 + v_rcp_f32 (avoid IEEE div expansion), park h in LDS
#pragma unroll
  for (int j = 0; j < 8; ++j) {
    float x = acc[j];
    float sig = __builtin_amdgcn_rcpf(1.f + __expf(-x));
    h_lds[(j + hi * 8) * H_STRIDE + ncol] = x * sig;
  }
  __syncthreads();

  // ---- GEMM2: o = h @ w2 + b2, K = 128 ----
  v8f acc2;
  {
    float bias = b2[ncol];
#pragma unroll
    for (int j = 0; j < 8; ++j) acc2[j] = bias;
  }
  const float* hrow = h_lds + l16 * H_STRIDE;
#pragma unroll 4
  for (int kt = 0; kt < HID / 4; ++kt) {
    int k = kt * 4 + hi * 2;
    v2f a, b;
    a.x = hrow[k];
    a.y = hrow[k + 1];
    b.x = w2[(size_t)k * D + ncol];
    b.y = w2[(size_t)(k + 1) * D + ncol];
    acc2 = __builtin_amdgcn_wmma_f32_16x16x4_f32(false, a, false, b,
                                                 (short)0, acc2, false, false);
  }
  // stash pre-norm output in LDS (reuse cat_lds)
  float* out_lds = cat_lds;
#pragma unroll
  for (int j = 0; j < 8; ++j)
    out_lds[(j + hi * 8) * H_STRIDE + ncol] = acc2[j];
  __syncthreads();

  // ---- LayerNorm + residual: 2 rows per wave, wave32 shfl reduction ----
#pragma unroll
  for (int rr = 0; rr < 2; ++rr) {
    int r = wave * 2 + rr;
    const float* row = out_lds + r * H_STRIDE;
    float4 x = *(const float4*)(row + lane * 4);
    float s  = x.x + x.y + x.z + x.w;
    float sq = x.x * x.x + x.y * x.y + x.z * x.z + x.w * x.w;
#pragma unroll
    for (int off = 16; off >= 1; off >>= 1) {
      s  += __shfl_xor(s, off, 32);
      sq += __shfl_xor(sq, off, 32);
    }
    float mean = s * (1.0f / D);
    float var  = sq * (1.0f / D) - mean * mean;
    float rstd = rsqrtf(var + 1e-5f);
    int node = tile0 + r;
    int c    = lane * 4;
    float4 g  = *(const float4*)(ln_g + c);
    float4 bb = *(const float4*)(ln_b + c);
    float4 nf = *(const float4*)(nfeat + (size_t)node * D + c);
    float4 o;
    o.x = (x.x - mean) * rstd * g.x + bb.x + nf.x;
    o.y = (x.y - mean) * rstd * g.y + bb.y + nf.y;
    o.z = (x.z - mean) * rstd * g.z + bb.z + nf.z;
    o.w = (x.w - mean) * rstd * g.w + bb.w + nf.w;
    *(float4*)(out + (size_t)node * D + c) = o;
  }
}

extern "C" void kernel_launch(void* const* d_in, const int* in_sizes, int n_in,
                              void* d_out, int out_size, void* d_ws, size_t ws_size,
                              hipStream_t stream) {
  const float* efeat   = (const float*)d_in[0];
  const float* nfeat   = (const float*)d_in[1];
  const int*   dst_idx = (const int*)d_in[2];
  const float* w1      = (const float*)d_in[3];
  const float* b1      = (const float*)d_in[4];
  const float* w2      = (const float*)d_in[5];
  const float* b2      = (const float*)d_in[6];
  const float* ln_g    = (const float*)d_in[7];
  const float* ln_b    = (const float*)d_in[8];
  float* out = (float*)d_out;
  float* agg = (float*)d_ws;      // N_NODES * D floats = 25.6 MB scratch

  // 1) zero the aggregation buffer
  {
    int n4 = (N_NODES * D) / 4;
    int blocks = (n4 + 255) / 256;
    zero_kernel<<<blocks, 256, 0, stream>>>((float4*)agg, n4);
  }
  // 2) segment-sum scatter (one wave32 per edge)
  {
    long long threads = (long long)N_EDGES * 32;
    int blocks = (int)((threads + 255) / 256);
    scatter_kernel<<<blocks, 256, 0, stream>>>(efeat, dst_idx, agg);
  }
  // 3) fused MLP + LayerNorm + residual (WMMA f32)
  {
    int blocks = N_NODES / TILE;  // 3125, exact
    mlp_kernel<<<blocks, 256, 0, stream>>>(agg, nfeat, w1, b1, w2, b2,
                                           ln_g, ln_b, out);
  }
}